// ScaledDotProductAttention_37151467110605
// MI455X (gfx1250) — compile-verified
//
#include <hip/hip_runtime.h>
#include <cstdint>

typedef __attribute__((ext_vector_type(2))) float v2f;
typedef __attribute__((ext_vector_type(8))) float v8f;

namespace {
constexpr int NB = 8;          // batch
constexpr int NH = 8;          // heads
constexpr int LL = 1024;       // sequence length
constexpr int DK = 32;         // head dim
constexpr int QT = 64;         // query rows per block
constexpr int SSTR = LL + 4;   // padded score row stride (floats) -> bank-conflict-free WMMA ld/st
constexpr int QSTR = DK + 2;   // padded Q row stride
}

__global__ __launch_bounds__(256, 1)
void sdpa_wmma_f32_kernel(const float* __restrict__ Q, const float* __restrict__ K,
                          const float* __restrict__ V, const uint8_t* __restrict__ Msk,
                          const float* __restrict__ R, float* __restrict__ ctx,
                          float* __restrict__ attn)
{
    __shared__ float sS[QT * SSTR];   // 64 x 1028 f32 = 257 KB score tile (CDNA5 320KB LDS)
    __shared__ float sQ[QT * QSTR];   // scaled Q tile
    __shared__ float sRed[QT * 4];
    __shared__ float sRow[QT];

    const int bh   = blockIdx.y;
    const int q0   = blockIdx.x * QT;
    const int tid  = threadIdx.x;
    const int lane = tid & 31;
    const int wave = tid >> 5;
    const int lmod = lane & 15;       // row (A) / col (B,D) within 16-wide tile
    const int lhalf = lane >> 4;      // 0: K=0..1 / M=0..7 ; 1: K=2..3 / M=8..15

    const float* Qb = Q + ((size_t)bh * LL + q0) * DK;
    const float* Kb = K + (size_t)bh * LL * DK;
    const float* Vb = V + (size_t)bh * LL * DK;
    const float* Rb = R + ((size_t)bh * LL + q0) * LL;
    const uint8_t* Mb = Msk + ((size_t)bh * LL + q0) * LL;
    float* Cb = ctx  + ((size_t)bh * LL + q0) * DK;
    float* Ab = attn + ((size_t)bh * LL + q0) * LL;

    const float scale = 0.17677669529663687f; // 1/sqrt(32)

    // stage Q tile into LDS, pre-scaled so S = (scale*Q)*K^T + res in one WMMA chain
    for (int i = tid; i < QT * DK; i += 256) {
        const int r = i >> 5, c = i & (DK - 1);
        sQ[r * QSTR + c] = Qb[r * DK + c] * scale;
    }
    __syncthreads();

    // -------- Phase 1: S tile (64 x 1024) via v_wmma_f32_16x16x4_f32, C preloaded with res_att
    {
        const int nt0 = wave * 8;                 // 8 n-tiles (128 key cols) per wave
        for (int j = 0; j < 8; ++j) {
            const int n0 = (nt0 + j) * 16;
            // B = K^T chunk: lane holds K[n0+lmod][4c + 2*lhalf + {0,1}] (contiguous pair)
            v2f b[8];
            const float* kp = Kb + (size_t)(n0 + lmod) * DK + 2 * lhalf;
            #pragma unroll
            for (int c = 0; c < 8; ++c) b[c] = *(const v2f*)(kp + 4 * c);

            v8f acc[4];
            #pragma unroll
            for (int mt = 0; mt < 4; ++mt) {
                #pragma unroll
                for (int v = 0; v < 8; ++v) {
                    const int m = mt * 16 + 8 * lhalf + v;
                    acc[mt][v] = Rb[(size_t)m * LL + n0 + lmod];   // residual bias as C
                }
            }
            #pragma unroll
            for (int c = 0; c < 8; ++c) {        // 4 independent WMMA chains interleaved
                #pragma unroll
                for (int mt = 0; mt < 4; ++mt) {
                    v2f a = *(const v2f*)(&sQ[(mt * 16 + lmod) * QSTR + 2 * lhalf + 4 * c]);
                    acc[mt] = __builtin_amdgcn_wmma_f32_16x16x4_f32(
                        false, a, false, b[c], (short)0, acc[mt], false, false);
                }
            }
            #pragma unroll
            for (int mt = 0; mt < 4; ++mt) {
                #pragma unroll
                for (int v = 0; v < 8; ++v) {
                    const int m = mt * 16 + 8 * lhalf + v;
                    const int n = n0 + lmod;
                    float s = acc[mt][v];
                    if (Mb[(size_t)m * LL + n]) s = -1e9f;
                    sS[m * SSTR + n] = s;
                }
            }
        }
    }
    __syncthreads();

    // -------- Phase 2: row softmax entirely in LDS (4 threads per row, 256-wide segments)
    const int row = tid & 63;
    const int seg = tid >> 6;
    float* sp = &sS[row * SSTR + seg * 256];

    float mx = -3.402823466e38f;
    for (int i = 0; i < 256; ++i) mx = fmaxf(mx, sp[i]);
    sRed[row * 4 + seg] = mx;
    __syncthreads();
    if (tid < QT)
        sRow[tid] = fmaxf(fmaxf(sRed[tid*4], sRed[tid*4+1]), fmaxf(sRed[tid*4+2], sRed[tid*4+3]));
    __syncthreads();
    const float rmax = sRow[row];

    float sum = 0.f;
    for (int i = 0; i < 256; ++i) {
        const float e = __expf(sp[i] - rmax);
        sp[i] = e;
        sum += e;
    }
    sRed[row * 4 + seg] = sum;
    __syncthreads();
    if (tid < QT)
        sRow[tid] = 1.0f / (sRed[tid*4] + sRed[tid*4+1] + sRed[tid*4+2] + sRed[tid*4+3]);
    __syncthreads();
    const float rinv = sRow[row];
    for (int i = 0; i < 256; ++i) sp[i] *= rinv;
    __syncthreads();

    // coalesced attn writeback
    for (int i = tid; i < QT * LL; i += 256) {
        const int r = i >> 10, c = i & (LL - 1);
        Ab[(size_t)r * LL + c] = sS[r * SSTR + c];
    }

    // -------- Phase 3: context = P (LDS) x V (L2-resident), one 16x16 tile per wave
    {
        const int mt = wave >> 1;
        const int n0 = (wave & 1) * 16;
        const float* vp = Vb + (size_t)(2 * lhalf) * DK + n0 + lmod;
        const float* pp = &sS[(mt * 16 + lmod) * SSTR + 2 * lhalf];
        v8f acc[4] = {v8f{}, v8f{}, v8f{}, v8f{}};  // 4-deep pipeline over K
        for (int k0 = 0; k0 < LL; k0 += 16) {
            #pragma unroll
            for (int u = 0; u < 4; ++u) {
                const int kk = k0 + 4 * u;
                v2f a = *(const v2f*)(pp + kk);
                const float* vv = vp + (size_t)kk * DK;
                v2f b; b[0] = vv[0]; b[1] = vv[DK];
                acc[u] = __builtin_amdgcn_wmma_f32_16x16x4_f32(
                    false, a, false, b, (short)0, acc[u], false, false);
            }
        }
        #pragma unroll
        for (int v = 0; v < 8; ++v) {
            const int m = mt * 16 + 8 * lhalf + v;
            Cb[(size_t)m * DK + n0 + lmod] = acc[0][v] + acc[1][v] + acc[2][v] + acc[3][v];
        }
    }
}

extern "C" void kernel_launch(void* const* d_in, const int* in_sizes, int n_in,
                              void* d_out, int out_size, void* d_ws, size_t ws_size,
                              hipStream_t stream) {
    (void)in_sizes; (void)n_in; (void)out_size; (void)d_ws; (void)ws_size;
    const float*   Q  = (const float*)d_in[0];
    const float*   K  = (const float*)d_in[1];
    const float*   V  = (const float*)d_in[2];
    const uint8_t* Mk = (const uint8_t*)d_in[3];   // jnp bool -> 1 byte
    const float*   R  = (const float*)d_in[4];
    float* ctx  = (float*)d_out;                                    // [B,H,L,DK]
    float* attn = ctx + (size_t)NB * NH * LL * DK;                  // [B,H,L,L]
    dim3 grid(LL / QT, NB * NH);
    sdpa_wmma_f32_kernel<<<grid, 256, 0, stream>>>(Q, K, V, Mk, R, ctx, attn);
}